// CoSparseGATLayer_19971597926796
// MI455X (gfx1250) — compile-verified
//
#include <hip/hip_runtime.h>
#include <math.h>

typedef __attribute__((ext_vector_type(2))) float v2f;
typedef __attribute__((ext_vector_type(8))) float v8f;

#define NHEADS 4
#define FOUT   32
#define FINCH  128   // FIN == NH*FOUT == 128

// ---------------------------------------------------------------------------
// Init: zero the output accumulator, softmax denominators, and max-keys.
// ---------------------------------------------------------------------------
__global__ void gat_init(float* __restrict__ out, float* __restrict__ denom,
                         unsigned* __restrict__ mkeys, int n_out, int n_den) {
  int i = blockIdx.x * blockDim.x + threadIdx.x;
  if (i < n_out) out[i] = 0.0f;
  if (i < n_den) denom[i] = 0.0f;
  if (i < 2)     mkeys[i] = 0u;
}

// Monotone float -> uint key (order-preserving) for atomicMax on floats.
__device__ __forceinline__ unsigned fkey(float f) {
  unsigned u = __float_as_uint(f);
  return (u & 0x80000000u) ? ~u : (u | 0x80000000u);
}
__device__ __forceinline__ float fdekey(unsigned u) {
  return __uint_as_float((u & 0x80000000u) ? (u & 0x7FFFFFFFu) : ~u);
}

__device__ __forceinline__ v8f wmma_f32(v2f a, v2f b, v8f c) {
  return __builtin_amdgcn_wmma_f32_16x16x4_f32(
      /*neg_a=*/false, a, /*neg_b=*/false, b,
      /*c_mod=*/(short)0, c, /*reuse_a=*/false, /*reuse_b=*/false);
}

// ---------------------------------------------------------------------------
// WMMA fp32 GEMM:  Y[n x 128] = X[n x 128] @ W[128 x 128]
// Block = 256 threads = 8 waves; block tile = 32(M) x 128(N).
// Each wave owns one 16-col N slice and TWO M-stacked 16x16 accumulators,
// so every B fragment (global load) feeds two V_WMMA_F32_16X16X4_F32 ops.
// A tile (32x128, 16 KB) staged in LDS; K-loop = 32 steps of K=4.
//
// VGPR layouts per CDNA5 ISA 7.12.2 (fp32):
//   A 16x4 : lanes 0-15 -> K={0,1} in v[0..1]; lanes 16-31 -> K={2,3}
//   B 4x16 : v0: lanes0-15 row K0, lanes16-31 row K2; v1: rows K1/K3
//   C 16x16: vj: lanes0-15 -> M=j, lanes16-31 -> M=j+8; N = lane&15
// ---------------------------------------------------------------------------
__global__ __launch_bounds__(256)
void gat_proj_wmma(const float* __restrict__ X, const float* __restrict__ W,
                   float* __restrict__ Y, int nrows) {
  __shared__ float As[32 * FINCH];  // 16 KB

  const int  m0   = blockIdx.x * 32;
  const int  tid  = threadIdx.x;
  const int  wave = tid >> 5;       // 0..7 -> which 16-col N slice
  const int  lane = tid & 31;
  const bool full = (m0 + 32 <= nrows);

  // Cooperative load of the 32x128 A tile (1024 float4 / 256 threads).
  {
    float4* dst = (float4*)As;
    const float4* src = (const float4*)(X + (size_t)m0 * FINCH);
    if (full) {
      #pragma unroll
      for (int i = 0; i < 4; ++i) dst[tid + i * 256] = src[tid + i * 256];
    } else {
      for (int i = tid; i < 32 * FINCH / 4; i += 256) {
        int row = i >> 5;  // 32 float4 per row
        float4 v = make_float4(0.f, 0.f, 0.f, 0.f);
        if (m0 + row < nrows) v = src[i];
        dst[i] = v;
      }
    }
  }
  __syncthreads();

  const int n0    = wave * 16;
  const int mrow  = lane & 15;
  const int khalf = lane >> 4;      // 0: K pair {0,1}; 1: K pair {2,3}

  v8f acc0 = {};   // rows m0      .. m0+15
  v8f acc1 = {};   // rows m0+16   .. m0+31
  #pragma unroll 4
  for (int k = 0; k < FINCH; k += 4) {
    v2f a0, a1, b;
    const float* ap = &As[mrow * FINCH + k + khalf * 2];
    a0.x = ap[0];
    a0.y = ap[1];
    a1.x = ap[16 * FINCH + 0];
    a1.y = ap[16 * FINCH + 1];
    const float* bp = W + (size_t)(k + khalf * 2) * FINCH + n0 + mrow;
    b.x = bp[0];
    b.y = bp[FINCH];
    acc0 = wmma_f32(a0, b, acc0);
    acc1 = wmma_f32(a1, b, acc1);
  }

  // Write-out: vgpr j -> row base + khalf*8 + j, col n0 + (lane&15)
  const int col = n0 + mrow;
  const int r0  = m0 + khalf * 8;
  if (full) {
    #pragma unroll
    for (int j = 0; j < 8; ++j) {
      Y[(size_t)(r0 + j)      * FINCH + col] = acc0[j];
      Y[(size_t)(r0 + 16 + j) * FINCH + col] = acc1[j];
    }
  } else {
    #pragma unroll
    for (int j = 0; j < 8; ++j) {
      int ra = r0 + j, rb = r0 + 16 + j;
      if (ra < nrows) Y[(size_t)ra * FINCH + col] = acc0[j];
      if (rb < nrows) Y[(size_t)rb * FINCH + col] = acc1[j];
    }
  }
}

// ---------------------------------------------------------------------------
// Per-node attention scores:  s[n,h] = sum_f proj[n,h*32+f] * a[h,f]
// plus a global (order-preserving-key) atomic max over all scores.
// ---------------------------------------------------------------------------
__global__ __launch_bounds__(256)
void gat_node_scores(const float* __restrict__ proj, const float* __restrict__ avec,
                     float* __restrict__ s, unsigned* __restrict__ mkey, int n) {
  int node = blockIdx.x * blockDim.x + threadIdx.x;
  if (node >= n) return;
  const float4* p = (const float4*)(proj + (size_t)node * FINCH);
  float m = -INFINITY;
  #pragma unroll
  for (int h = 0; h < NHEADS; ++h) {
    const float4* av = (const float4*)(avec + h * FOUT);
    float acc = 0.0f;
    #pragma unroll
    for (int i = 0; i < FOUT / 4; ++i) {
      float4 x = p[h * (FOUT / 4) + i];
      float4 y = av[i];
      acc += x.x * y.x + x.y * y.y + x.z * y.z + x.w * y.w;
    }
    s[node * NHEADS + h] = acc;
    m = fmaxf(m, acc);
  }
  atomicMax(mkey, fkey(m));
}

// Combine the two per-side maxima into the softmax shift constant C.
// (Any constant shift leaves softmax invariant; leaky_relu(ms+mt) is an
//  upper bound on the true per-edge max -> exp args <= 0, overflow-safe.)
__global__ void gat_combine_max(const unsigned* __restrict__ mkeys,
                                float* __restrict__ C) {
  if (threadIdx.x == 0 && blockIdx.x == 0) {
    float x = fdekey(mkeys[0]) + fdekey(mkeys[1]);
    C[0] = (x > 0.0f) ? x : 0.2f * x;
  }
}

// ---------------------------------------------------------------------------
// Edge pass 1: numerator w = exp(leaky_relu(s_src[si]+s_trg[ti]) - C),
// scatter-add into per-target denominator. One thread per (edge, head).
// ---------------------------------------------------------------------------
__global__ __launch_bounds__(256)
void gat_edge_softmax(const int* __restrict__ si, const int* __restrict__ ti,
                      const float* __restrict__ s_src, const float* __restrict__ s_trg,
                      const float* __restrict__ C, float* __restrict__ edge_w,
                      float* __restrict__ denom, int e4) {
  int idx = blockIdx.x * blockDim.x + threadIdx.x;
  if (idx >= e4) return;
  int e = idx >> 2;
  int h = idx & 3;
  int s = si[e];
  int t = ti[e];
  float x = s_src[s * NHEADS + h] + s_trg[t * NHEADS + h];
  x = (x > 0.0f) ? x : 0.2f * x;
  float w = __expf(x - C[0]);
  edge_w[idx] = w;
  atomicAdd(&denom[t * NHEADS + h], w);
}

// ---------------------------------------------------------------------------
// Edge pass 2: out[ti] += src_proj[si] * att. One wave32 per edge; each lane
// handles 4 consecutive features (coalesced float4 gather, 4 f32 atomics).
// src_proj (25.6 MB) and out (25.6 MB) both L2-resident on MI455X (192 MB).
// ---------------------------------------------------------------------------
__global__ __launch_bounds__(256)
void gat_edge_aggregate(const int* __restrict__ si, const int* __restrict__ ti,
                        const float* __restrict__ src_proj,
                        const float* __restrict__ edge_w,
                        const float* __restrict__ denom,
                        float* __restrict__ out, int E) {
  int lane = threadIdx.x & 31;
  int eid  = (blockIdx.x * blockDim.x + threadIdx.x) >> 5;
  if (eid >= E) return;
  int s = si[eid];
  int t = ti[eid];
  int f0 = lane * 4;          // 0..124
  int h  = f0 >> 5;           // head = feature block / 32
  float att = edge_w[eid * NHEADS + h] / (denom[t * NHEADS + h] + 1e-16f);
  float4 v = *(const float4*)(src_proj + (size_t)s * FINCH + f0);
  float* o = out + (size_t)t * FINCH + f0;
  atomicAdd(o + 0, v.x * att);
  atomicAdd(o + 1, v.y * att);
  atomicAdd(o + 2, v.z * att);
  atomicAdd(o + 3, v.w * att);
}

// ---------------------------------------------------------------------------
// Host-side launch
// ---------------------------------------------------------------------------
extern "C" void kernel_launch(void* const* d_in, const int* in_sizes, int n_in,
                              void* d_out, int out_size, void* d_ws, size_t ws_size,
                              hipStream_t stream) {
  const float* trg   = (const float*)d_in[0];
  const float* src   = (const float*)d_in[1];
  const int*   eidx  = (const int*)d_in[2];
  const float* Wt    = (const float*)d_in[3];
  const float* Ws    = (const float*)d_in[4];
  const float* a_src = (const float*)d_in[5];
  const float* a_trg = (const float*)d_in[6];
  float* out = (float*)d_out;

  const int NT = in_sizes[0] / FINCH;
  const int NS = in_sizes[1] / FINCH;
  const int E  = in_sizes[2] / 2;
  const int* si = eidx;       // edge_index[0]
  const int* ti = eidx + E;   // edge_index[1]

  // Workspace carve-up (floats). trg_proj is dead after node-scores, so the
  // edge numerator buffer (E*4 <= NT*128 floats) aliases it.
  float* ws       = (float*)d_ws;
  float* src_proj = ws;                               // NS*128
  float* scratch  = src_proj + (size_t)NS * FINCH;    // max(NT*128, E*4)
  float* trg_proj = scratch;
  float* edge_w   = scratch;
  float* s_src    = scratch + (size_t)NT * FINCH;     // NS*4
  float* s_trg    = s_src + (size_t)NS * NHEADS;      // NT*4
  float* denom    = s_trg + (size_t)NT * NHEADS;      // NT*4
  unsigned* mkeys = (unsigned*)(denom + (size_t)NT * NHEADS);  // 2
  float* Cval     = (float*)(mkeys + 2);                       // 1

  // 1) zero out + denom + max keys
  {
    int n_out = NT * FINCH;
    int n_den = NT * NHEADS;
    int blocks = (n_out + 255) / 256;
    gat_init<<<blocks, 256, 0, stream>>>(out, denom, mkeys, n_out, n_den);
  }
  // 2) WMMA projections (32-row tiles)
  gat_proj_wmma<<<(NS + 31) / 32, 256, 0, stream>>>(src, Ws, src_proj, NS);
  gat_proj_wmma<<<(NT + 31) / 32, 256, 0, stream>>>(trg, Wt, trg_proj, NT);
  // 3) per-node scores + per-side maxima
  gat_node_scores<<<(NS + 255) / 256, 256, 0, stream>>>(src_proj, a_src, s_src, mkeys + 0, NS);
  gat_node_scores<<<(NT + 255) / 256, 256, 0, stream>>>(trg_proj, a_trg, s_trg, mkeys + 1, NT);
  // 4) softmax shift constant
  gat_combine_max<<<1, 32, 0, stream>>>(mkeys, Cval);
  // 5) edge softmax numerators + denominators
  {
    int e4 = E * NHEADS;
    gat_edge_softmax<<<(e4 + 255) / 256, 256, 0, stream>>>(si, ti, s_src, s_trg,
                                                           Cval, edge_w, denom, e4);
  }
  // 6) weighted aggregation (1 wave per edge)
  {
    long long threads = (long long)E * 32;
    int blocks = (int)((threads + 255) / 256);
    gat_edge_aggregate<<<blocks, 256, 0, stream>>>(si, ti, src_proj, edge_w,
                                                   denom, out, E);
  }
}